// DesTweetConsistency_11484742549567
// MI455X (gfx1250) — compile-verified
//
#include <hip/hip_runtime.h>
#include <math.h>

// Problem constants (reference: B=1024, T=200, F=768, H=768)
#define B_ 1024
#define T_ 200
#define F_ 768
#define H_ 768

typedef float v2f __attribute__((ext_vector_type(2)));
typedef float v8f __attribute__((ext_vector_type(8)));

// ---------------------------------------------------------------------------
// Generic fp32 WMMA GEMM:  C[M,N] = A[M,K] @ B[K,N] (+ bias[n]*biasScale[m])
// optionally leaky-relu.  transB=1 reads B as B[n,k] (i.e. multiplies by B^T).
// Requires M % 64 == 0 and N % 64 == 0 (true for all call sites; K arbitrary).
// Block tile 64x64, 256 threads = 8 waves arranged 4(M) x 2(N); each wave
// owns a 16x32 tile => two v_wmma_f32_16x16x4_f32 accumulators, K-step 4.
// LDS tiles stored so each lane's K-pair is contiguous -> single ds_load_b64
// per fragment, no repacking movs.  Full K-tiles use b128 global loads and
// b128 LDS stores with a uniform (scalar) branch; only the K-remainder tile
// takes the guarded scalar path.
// ---------------------------------------------------------------------------
__global__ __launch_bounds__(256)
void gemm_f32_wmma(const float* __restrict__ A, const float* __restrict__ Bm,
                   const float* __restrict__ bias,
                   const float* __restrict__ biasScale,
                   float* __restrict__ C,
                   int M, int N, int K, int transB, int relu)
{
    __shared__ float As[64][20];   // [m][k], row stride 80B (16B aligned)
    __shared__ float Bs[64][20];   // [n][k] (B tile stored transposed)

    const int tid  = threadIdx.x;
    const int lane = tid & 31;
    const int wid  = tid >> 5;
    const int wm   = wid & 3;          // wave row (0..3) -> 16 rows each
    const int wn   = wid >> 2;         // wave col (0..1) -> 32 cols each
    const int hi   = lane >> 4;        // half-wave: K-pair select / M+8
    const int lh   = lane & 15;

    const int m0 = blockIdx.y * 64;
    const int n0 = blockIdx.x * 64;

    v8f acc0 = {0.f,0.f,0.f,0.f,0.f,0.f,0.f,0.f};
    v8f acc1 = {0.f,0.f,0.f,0.f,0.f,0.f,0.f,0.f};

    for (int k0 = 0; k0 < K; k0 += 16) {
        const bool full = (k0 + 16 <= K);     // uniform across block
        if (full) {
            // A tile: 64 rows x 16 k; thread -> (row = tid/4, 4 k's)
            {
                const int r = tid >> 2, c = (tid & 3) << 2;
                const float4 av =
                    *(const float4*)&A[(size_t)(m0 + r) * K + k0 + c];
                *(float4*)&As[r][c] = av;
            }
            if (transB) {
                // B^T: rows of Bm are already [n][k] -> straight b128 copy
                const int n = tid >> 2, c = (tid & 3) << 2;
                const float4 bv =
                    *(const float4*)&Bm[(size_t)(n0 + n) * K + k0 + c];
                *(float4*)&Bs[n][c] = bv;
            } else {
                // B: [k][n] in memory -> transpose into Bs[n][k]
                const int r = tid >> 4, c4 = (tid & 15) << 2;
                const float4 bv =
                    *(const float4*)&Bm[(size_t)(k0 + r) * N + n0 + c4];
                Bs[c4 + 0][r] = bv.x;
                Bs[c4 + 1][r] = bv.y;
                Bs[c4 + 2][r] = bv.z;
                Bs[c4 + 3][r] = bv.w;
            }
        } else {
            // K-remainder tile (only for K=200 GEMM): guarded, zero-padded
            #pragma unroll
            for (int i = 0; i < 4; ++i) {
                int idx = tid + i * 256;
                int r = idx >> 4, c = idx & 15;
                As[r][c] = (k0 + c < K) ? A[(size_t)(m0 + r) * K + k0 + c]
                                        : 0.0f;
            }
            #pragma unroll
            for (int i = 0; i < 4; ++i) {
                int idx = tid + i * 256;
                float val = 0.0f;
                int n, c;
                if (transB) { n = idx >> 4;  c = idx & 15; }
                else        { n = idx & 63;  c = idx >> 6; }
                if (k0 + c < K)
                    val = transB ? Bm[(size_t)(n0 + n) * K + k0 + c]
                                 : Bm[(size_t)(k0 + c) * N + n0 + n];
                Bs[n][c] = val;
            }
        }
        __syncthreads();

        #pragma unroll
        for (int kb = 0; kb < 16; kb += 4) {
            const v2f a  = *(const v2f*)&As[wm * 16 + lh][kb + 2 * hi];
            const v2f b0 = *(const v2f*)&Bs[wn * 32 + lh][kb + 2 * hi];
            const v2f b1 = *(const v2f*)&Bs[wn * 32 + 16 + lh][kb + 2 * hi];
            acc0 = __builtin_amdgcn_wmma_f32_16x16x4_f32(
                       false, a, false, b0, (short)0, acc0, false, false);
            acc1 = __builtin_amdgcn_wmma_f32_16x16x4_f32(
                       false, a, false, b1, (short)0, acc1, false, false);
        }
        __syncthreads();
    }

    // Epilogue. D layout: VGPR r -> row (wm*16 + hi*8 + r), col (wn*32 + lh)
    const int rowBase = m0 + wm * 16 + hi * 8;
    const int colA = n0 + wn * 32 + lh;
    const int colB = colA + 16;
    const float bc0 = bias ? bias[colA] : 0.0f;
    const float bc1 = bias ? bias[colB] : 0.0f;
    #pragma unroll
    for (int r = 0; r < 8; ++r) {
        const int row = rowBase + r;
        const float bs = biasScale ? biasScale[row] : 1.0f;
        float v0 = acc0[r] + bc0 * bs;
        float v1 = acc1[r] + bc1 * bs;
        if (relu) {
            v0 = (v0 >= 0.0f) ? v0 : 0.01f * v0;
            v1 = (v1 >= 0.0f) ? v1 : 0.01f * v1;
        }
        C[(size_t)row * N + colA] = v0;
        C[(size_t)row * N + colB] = v1;
    }
}

// ---------------------------------------------------------------------------
// Pass 1 over tweets: score[b,t] = tweets[b,t,:] . v[b,:] + sim_b . des_sim[b,:]
// One block per batch row; v[b] cached in LDS as float4; one wave per tweet;
// 6 x global_load_b128 per tweet-dot to run at HBM line rate.
// ---------------------------------------------------------------------------
__global__ __launch_bounds__(256)
void score_kernel(const float* __restrict__ tweets,
                  const float* __restrict__ v,
                  const float* __restrict__ des_sim,
                  const float* __restrict__ sim_b,
                  float* __restrict__ score)
{
    __shared__ float4 vs4[F_ / 4];      // 192 float4 = 3 KB
    __shared__ float red[8];
    const int b = blockIdx.x;
    const int tid = threadIdx.x;
    const int lane = tid & 31;
    const int wid  = tid >> 5;

    if (tid < F_ / 4)
        vs4[tid] = ((const float4*)(v + (size_t)b * F_))[tid];

    // cb = sim_b . des_sim[b]
    float p = 0.0f;
    for (int f = tid; f < F_; f += 256)
        p += sim_b[f] * des_sim[(size_t)b * F_ + f];
    #pragma unroll
    for (int o = 16; o > 0; o >>= 1) p += __shfl_xor(p, o, 32);
    if (lane == 0) red[wid] = p;
    __syncthreads();
    float cb = 0.0f;
    #pragma unroll
    for (int i = 0; i < 8; ++i) cb += red[i];

    const float* tw = tweets + (size_t)b * T_ * F_;
    for (int t = wid; t < T_; t += 8) {
        const float4* row4 = (const float4*)(tw + (size_t)t * F_);
        if (t + 8 < T_)  // prefetch this wave's next tweet row
            __builtin_prefetch(tw + (size_t)(t + 8) * F_ + 4 * lane, 0, 0);
        float d = 0.0f;
        #pragma unroll
        for (int i = 0; i < F_ / 128; ++i) {   // 6 iterations
            const float4 x = row4[lane + 32 * i];
            const float4 y = vs4[lane + 32 * i];
            d += x.x * y.x + x.y * y.y + x.z * y.z + x.w * y.w;
        }
        #pragma unroll
        for (int o = 16; o > 0; o >>= 1) d += __shfl_xor(d, o, 32);
        if (lane == 0) score[b * T_ + t] = d + cb;
    }
}

// ---------------------------------------------------------------------------
// Per-batch: min-max normalize score, softmax(-w), l_weights = weights @ WM,
// s[b] = sum(l_weights[b,:]).  One 256-thread block per batch row.
// ---------------------------------------------------------------------------
__global__ __launch_bounds__(256)
void softmax_mix_kernel(const float* __restrict__ score,
                        const float* __restrict__ WM,
                        float* __restrict__ weights,
                        float* __restrict__ l_weights,
                        float* __restrict__ s_out)
{
    __shared__ float sc[T_];
    __shared__ float wt[T_];
    __shared__ float red[256];
    const int b = blockIdx.x;
    const int tid = threadIdx.x;

    if (tid < T_) sc[tid] = score[b * T_ + tid];
    __syncthreads();

    // min
    red[tid] = (tid < T_) ? sc[tid] : 3.402823466e+38f;
    __syncthreads();
    for (int o = 128; o > 0; o >>= 1) {
        if (tid < o) red[tid] = fminf(red[tid], red[tid + o]);
        __syncthreads();
    }
    const float mn = red[0];
    __syncthreads();
    // max
    red[tid] = (tid < T_) ? sc[tid] : -3.402823466e+38f;
    __syncthreads();
    for (int o = 128; o > 0; o >>= 1) {
        if (tid < o) red[tid] = fmaxf(red[tid], red[tid + o]);
        __syncthreads();
    }
    const float mx = red[0];
    __syncthreads();

    const float denom = mx - mn;
    float w = 0.0f;
    if (tid < T_) w = (denom != 0.0f) ? (sc[tid] - mn) / denom : 0.0f;

    // softmax(-w): max of (-w)
    red[tid] = (tid < T_) ? -w : -3.402823466e+38f;
    __syncthreads();
    for (int o = 128; o > 0; o >>= 1) {
        if (tid < o) red[tid] = fmaxf(red[tid], red[tid + o]);
        __syncthreads();
    }
    const float m2 = red[0];
    __syncthreads();

    float e = (tid < T_) ? __expf(-w - m2) : 0.0f;
    red[tid] = e;
    __syncthreads();
    for (int o = 128; o > 0; o >>= 1) {
        if (tid < o) red[tid] += red[tid + o];
        __syncthreads();
    }
    const float inv = 1.0f / red[0];
    __syncthreads();

    if (tid < T_) {
        const float wv = e * inv;
        wt[tid] = wv;
        weights[b * T_ + tid] = wv;
    }
    __syncthreads();

    // l_weights[b,t'] = sum_t wt[t] * WM[t,t']   (WM stays L2-resident)
    float lw = 0.0f;
    if (tid < T_) {
        #pragma unroll 4
        for (int t = 0; t < T_; ++t) lw += wt[t] * WM[t * T_ + tid];
        l_weights[b * T_ + tid] = lw;
    }
    red[tid] = (tid < T_) ? lw : 0.0f;
    __syncthreads();
    for (int o = 128; o > 0; o >>= 1) {
        if (tid < o) red[tid] += red[tid + o];
        __syncthreads();
    }
    if (tid == 0) s_out[b] = red[0];
}

// ---------------------------------------------------------------------------
// Pass 2 over tweets: pooled[b,f] = sum_t l_weights[b,t] * tweets[b,t,f]
// 192 threads/block; each thread owns one float4 column -> b128 loads/stores.
// ---------------------------------------------------------------------------
__global__ __launch_bounds__(192)
void pooled_kernel(const float* __restrict__ tweets,
                   const float* __restrict__ l_weights,
                   float* __restrict__ pooled)
{
    __shared__ float lw[T_];
    const int b = blockIdx.x;
    const int tid = threadIdx.x;
    for (int t = tid; t < T_; t += 192) lw[t] = l_weights[b * T_ + t];
    __syncthreads();

    float ax = 0.f, ay = 0.f, az = 0.f, aw = 0.f;
    const float* tw = tweets + (size_t)b * T_ * F_;
    for (int t = 0; t < T_; ++t) {
        const float w = lw[t];
        const float4 x = ((const float4*)(tw + (size_t)t * F_))[tid];
        if (t + 2 < T_)
            __builtin_prefetch(tw + (size_t)(t + 2) * F_ + 4 * tid, 0, 0);
        ax += w * x.x; ay += w * x.y; az += w * x.z; aw += w * x.w;
    }
    float4 acc; acc.x = ax; acc.y = ay; acc.z = az; acc.w = aw;
    ((float4*)(pooled + (size_t)b * F_))[tid] = acc;
}

// ---------------------------------------------------------------------------
extern "C" void kernel_launch(void* const* d_in, const int* in_sizes, int n_in,
                              void* d_out, int out_size, void* d_ws, size_t ws_size,
                              hipStream_t stream)
{
    const float* des    = (const float*)d_in[0];   // [B,F]
    const float* tweets = (const float*)d_in[1];   // [B,T,F]
    const float* WM     = (const float*)d_in[2];   // [T,T]
    const float* sim_w  = (const float*)d_in[3];   // [F,F]
    const float* sim_b  = (const float*)d_in[4];   // [F]
    const float* ws_w   = (const float*)d_in[5];   // [F,F]
    const float* ws_b   = (const float*)d_in[6];   // [F]
    const float* desp_w = (const float*)d_in[7];   // [F,H]
    const float* desp_b = (const float*)d_in[8];   // [H]
    const float* wp_w   = (const float*)d_in[9];   // [T,H]
    const float* wp_b   = (const float*)d_in[10];  // [H]
    const float* wtp_w  = (const float*)d_in[11];  // [F,H]
    const float* wtp_b  = (const float*)d_in[12];  // [H]

    float* out      = (float*)d_out;
    float* des_out  = out;                         // [B,H]
    float* wgt_out  = out + (size_t)B_ * H_;       // [B,H]
    float* wt_out   = out + (size_t)2 * B_ * H_;   // [B,H]

    // workspace layout (floats): ~15 MB total
    float* ws       = (float*)d_ws;
    float* des_sim  = ws;                          // B*F
    float* vvec     = des_sim + (size_t)B_ * F_;   // B*F  (des_sim @ sim_w^T)
    float* pooled   = vvec    + (size_t)B_ * F_;   // B*F
    float* wtx      = pooled  + (size_t)B_ * F_;   // B*F  (weighted_tweets)
    float* score    = wtx     + (size_t)B_ * F_;   // B*T
    float* weights  = score   + (size_t)B_ * T_;   // B*T
    float* l_w      = weights + (size_t)B_ * T_;   // B*T
    float* s_b      = l_w     + (size_t)B_ * T_;   // B

    dim3 blk(256);
    dim3 gFH(H_ / 64, B_ / 64);   // N=768, M=1024

    // 1. des_sim = des @ sim_w + sim_b
    gemm_f32_wmma<<<gFH, blk, 0, stream>>>(des, sim_w, sim_b, nullptr,
                                           des_sim, B_, F_, F_, 0, 0);
    // 2. v = des_sim @ sim_w^T   (so score = tweets . v + sim_b . des_sim)
    gemm_f32_wmma<<<gFH, blk, 0, stream>>>(des_sim, sim_w, nullptr, nullptr,
                                           vvec, B_, F_, F_, 1, 0);
    // 3. streaming pass 1 over tweets -> score
    score_kernel<<<dim3(B_), blk, 0, stream>>>(tweets, vvec, des_sim, sim_b, score);
    // 4. normalize + softmax + weight mixing + row sums
    softmax_mix_kernel<<<dim3(B_), blk, 0, stream>>>(score, WM, weights, l_w, s_b);
    // 5. streaming pass 2 over tweets -> pooled
    pooled_kernel<<<dim3(B_), dim3(192), 0, stream>>>(tweets, l_w, pooled);
    // 6. weighted_tweets = pooled @ ws_w + s[b]*ws_b
    gemm_f32_wmma<<<gFH, blk, 0, stream>>>(pooled, ws_w, ws_b, s_b,
                                           wtx, B_, F_, F_, 0, 0);
    // 7. des_out = leaky_relu(des @ desp_w + desp_b)
    gemm_f32_wmma<<<gFH, blk, 0, stream>>>(des, desp_w, desp_b, nullptr,
                                           des_out, B_, H_, F_, 0, 1);
    // 8. weights_out = leaky_relu(weights @ wp_w + wp_b)  (K=200, padded tile)
    gemm_f32_wmma<<<gFH, blk, 0, stream>>>(weights, wp_w, wp_b, nullptr,
                                           wgt_out, B_, H_, T_, 0, 1);
    // 9. wt_out = leaky_relu(weighted_tweets @ wtp_w + wtp_b)
    gemm_f32_wmma<<<gFH, blk, 0, stream>>>(wtx, wtp_w, wtp_b, nullptr,
                                           wt_out, B_, H_, F_, 0, 1);
}